// UmapLoss_54365696032925
// MI455X (gfx1250) — compile-verified
//
#include <hip/hip_runtime.h>
#include <math.h>

#define NEG_RATE 5
#define EPS_F 1e-4f
#define PERM_MUL 1000003ull
#define PERM_ADD 12345ull

typedef __attribute__((ext_vector_type(2))) float v2f;
typedef __attribute__((ext_vector_type(8))) float v8f;

// Full-wave32 sum using V_WMMA_F32_16X16X4_F32 as a reduction engine.
// A (16x4 f32): vgpr0 = per-lane value v (lanes0-15 -> A[m][0], lanes16-31 -> A[m][2]),
// vgpr1 = 0. B = all ones. D[m][n] = v_m + v_{m+16} (replicated over n).
// Summing the 8 D vgprs in-lane gives lanes0-15 = sum(lanes 0-7,16-23),
// lanes16-31 = sum(lanes 8-15,24-31); one shfl_xor(16) completes the wave sum.
// EXEC is all-1s at every call site (reductions sit in uniform control flow).
__device__ __forceinline__ float wave_sum32(float v) {
  v2f a; a[0] = v;    a[1] = 0.0f;
  v2f b; b[0] = 1.0f; b[1] = 1.0f;
  v8f c = {};
  v8f d = __builtin_amdgcn_wmma_f32_16x16x4_f32(false, a, false, b,
                                                (short)0, c, false, false);
  float s = d[0] + d[1] + d[2] + d[3] + d[4] + d[5] + d[6] + d[7];
  s += __shfl_xor(s, 16, 32);
  return s;
}

__device__ __forceinline__ void row_stats(const float* __restrict__ p, int C,
                                          int& amax, float& conf) {
  float m = p[0]; int am = 0;
  for (int c = 1; c < C; ++c) { float v = p[c]; if (v > m) { m = v; am = c; } }
  float s = 0.0f;
  for (int c = 0; c < C; ++c) s += expf(p[c] - m);
  amax = am;
  conf = 1.0f / s;   // max softmax prob = exp(0)/sum
}

__global__ void k_init(float* acc) {
  if (threadIdx.x < 16) acc[threadIdx.x] = 0.0f;
}

// Per-row pass: pos_d, ce_pos, class/conf metadata, masked pos_d for margin pass.
__global__ void k_rows(const float* __restrict__ eto, const float* __restrict__ efr,
                       const float* __restrict__ probs,
                       const float* __restrict__ pto, const float* __restrict__ pfr,
                       float* __restrict__ acc,
                       float* __restrict__ conf_to, float* __restrict__ conf_from,
                       int* __restrict__ cls_pk, float* __restrict__ pm,
                       int B, int C, int D) {
  int i = blockIdx.x * blockDim.x + threadIdx.x;
  float pos_d = 0.0f, ce = 0.0f;
  if (i < B) {
    int cto, cfr; float cf_to, cf_fr;
    row_stats(pto + (long long)i * C, C, cto, cf_to);
    row_stats(pfr + (long long)i * C, C, cfr, cf_fr);

    float d2 = 0.0f;
    for (int k = 0; k < D; ++k) {
      float dx = eto[(long long)i * D + k] - efr[(long long)i * D + k];
      d2 += dx * dx;
    }
    pos_d = sqrtf(d2);

    float p = 1.0f / (1.0f + d2);            // prob_pos (A=1, B=1)
    float q = d2 / (1.0f + d2);              // 1 - prob_pos
    p = fminf(fmaxf(p, EPS_F), 1.0f);
    q = fminf(fmaxf(q, EPS_F), 1.0f);
    float pr = probs[i];
    ce = -pr * logf(p) - (1.0f - pr) * logf(q);   // REPULSION = 1

    conf_to[i]   = cf_to;
    conf_from[i] = cf_fr;
    cls_pk[i]    = (cto & 0xFF) | ((cfr & 0xFF) << 8);
    pm[i]        = (cto == cfr) ? -1.0f : pos_d;  // relu(margin - d) only when !same
  }
  float s0 = wave_sum32(pos_d);
  float s1 = wave_sum32(ce);
  if ((threadIdx.x & 31) == 0) {
    atomicAdd(&acc[0], s0);   // sum pos_d
    atomicAdd(&acc[1], s1);   // sum ce_pos
  }
}

// Negative-pair pass: grid-stride over n_neg, register accumulation,
// WMMA wave reduction once at the end.
__global__ void k_neg(const float* __restrict__ eto, const float* __restrict__ efr,
                      const float* __restrict__ conf_to, const float* __restrict__ conf_from,
                      const int* __restrict__ cls_pk,
                      float* __restrict__ acc, int B, int D) {
  const long long n_neg = (long long)B * NEG_RATE;
  const long long stride = (long long)gridDim.x * blockDim.x;
  float sd = 0.0f, sn = 0.0f, sc = 0.0f;
  for (long long j = (long long)blockIdx.x * blockDim.x + threadIdx.x;
       j < n_neg; j += stride) {
    unsigned long long p = ((unsigned long long)j * PERM_MUL + PERM_ADD)
                           % (unsigned long long)n_neg;      // fixed bijection
    int ito = (int)(j / NEG_RATE);
    int ifr = (int)(p / NEG_RATE);

    int cto = cls_pk[ito] & 0xFF;
    int cfr = (cls_pk[ifr] >> 8) & 0xFF;
    float keep = ((cto == cfr) && (conf_from[ifr] == conf_to[ito])) ? 0.0f : 1.0f;

    float d2 = 0.0f;
    for (int k = 0; k < D; ++k) {
      float dx = eto[(long long)ito * D + k] - efr[(long long)ifr * D + k];
      d2 += dx * dx;
    }
    float d = sqrtf(d2);
    float q = d2 / (1.0f + d2);                 // 1 - prob_neg
    q = fminf(fmaxf(q, EPS_F), 1.0f);
    float ce = -logf(q);

    sd += d * keep;
    sn += keep;
    sc += ce * keep;
  }
  float a = wave_sum32(sd);
  float b = wave_sum32(sn);
  float c = wave_sum32(sc);
  if ((threadIdx.x & 31) == 0) {
    atomicAdd(&acc[2], a);   // sum neg_d * keep
    atomicAdd(&acc[3], b);   // neg_num
    atomicAdd(&acc[4], c);   // sum ce_neg * keep
  }
}

// Margin pass: batch_margin is now known from acc[0..3].
__global__ void k_margin(const float* __restrict__ pm, float* __restrict__ acc, int B) {
  float pos_mean = acc[0] / (float)B;
  float neg_mean = acc[2] / acc[3];
  float margin = 0.5f * (pos_mean + neg_mean);
  int i = blockIdx.x * blockDim.x + threadIdx.x;
  float m = 0.0f;
  if (i < B) {
    float v = pm[i];
    if (v >= 0.0f) m = fmaxf(margin - v, 0.0f);
  }
  float s = wave_sum32(m);
  if ((threadIdx.x & 31) == 0) atomicAdd(&acc[5], s);
}

__global__ void k_final(const float* __restrict__ acc, float* __restrict__ out, int B) {
  float neg_num = acc[3];
  float ce_mean = (acc[1] + acc[4]) / ((float)B + neg_num);
  float margin_loss = acc[5] / (float)B;
  out[0] = ce_mean + margin_loss;
}

extern "C" void kernel_launch(void* const* d_in, const int* in_sizes, int n_in,
                              void* d_out, int out_size, void* d_ws, size_t ws_size,
                              hipStream_t stream) {
  const float* eto   = (const float*)d_in[0];
  const float* efr   = (const float*)d_in[1];
  const float* probs = (const float*)d_in[2];
  const float* pto   = (const float*)d_in[3];
  const float* pfr   = (const float*)d_in[4];

  const int B = in_sizes[2];              // probs is [B]
  const int D = in_sizes[0] / B;          // 2
  const int C = in_sizes[3] / B;          // 10

  float* ws        = (float*)d_ws;
  float* acc       = ws;                  // 16 floats of accumulators
  float* conf_to   = ws + 16;             // [B]
  float* conf_from = conf_to + B;         // [B]
  float* pm        = conf_from + B;       // [B]
  int*   cls_pk    = (int*)(pm + B);      // [B]

  const int blk = 256;
  const int rowGrid = (B + blk - 1) / blk;

  k_init<<<1, 32, 0, stream>>>(acc);
  k_rows<<<rowGrid, blk, 0, stream>>>(eto, efr, probs, pto, pfr, acc,
                                      conf_to, conf_from, cls_pk, pm, B, C, D);
  k_neg<<<2048, blk, 0, stream>>>(eto, efr, conf_to, conf_from, cls_pk, acc, B, D);
  k_margin<<<rowGrid, blk, 0, stream>>>(pm, acc, B);
  k_final<<<1, 1, 0, stream>>>(acc, (float*)d_out, B);
}